// GatedLayer_45303315038878
// MI455X (gfx1250) — compile-verified
//
#include <hip/hip_runtime.h>
#include <hip/hip_bf16.h>

// -----------------------------------------------------------------------------
// Gated GNN layer for MI455X (gfx1250, wave32).
// Memory-bound scatter/gather workload (~2.2 GB moved, ~100us at 23.3 TB/s).
// No matmul exists in this op, so no WMMA; the CDNA5-specific wins are:
//   * wave32 == C(=32): one lane per class, reductions are shfl_xor butterflies
//   * one wave per edge: coalesced 128B probs rows / 512B h rows, contiguous
//     global_atomic_add_f32 bursts into L2 atomic units
//   * global_atomic_add_f64 for the whole-graph LayerNorm statistics
//   * global_prefetch_b8 (__builtin_prefetch) with 2-edges-per-wave software
//     pipelining to hide random-gather latency
//   * fused wave-per-node gating + epilogue (z computed in-wave, no gate array)
// -----------------------------------------------------------------------------

#define LN_EPS 1e-5f
#define LOG_EPS 1e-9f

__device__ __forceinline__ float wave_sum32(float v) {
    #pragma unroll
    for (int off = 16; off > 0; off >>= 1) v += __shfl_xor(v, off, 32);
    return v;
}
__device__ __forceinline__ float wave_max32(float v) {
    #pragma unroll
    for (int off = 16; off > 0; off >>= 1) v = fmaxf(v, __shfl_xor(v, off, 32));
    return v;
}

// ---- 0) zero the accumulation region of the workspace (float4 stores) -------
__global__ void k_zero(float4* __restrict__ p, int n4) {
    int i = blockIdx.x * blockDim.x + threadIdx.x;
    if (i < n4) p[i] = make_float4(0.f, 0.f, 0.f, 0.f);
}

// ---- 1) softmax: one wave per node, one lane per class (C == 32) ------------
__global__ void k_softmax(const float* __restrict__ logits,
                          float* __restrict__ probs, int N) {
    int wid  = (blockIdx.x * blockDim.x + threadIdx.x) >> 5;
    int lane = threadIdx.x & 31;
    if (wid >= N) return;
    size_t idx = (size_t)wid * 32 + lane;
    float x = logits[idx];
    float m = wave_max32(x);
    float e = __expf(x - m);
    float s = wave_sum32(e);
    probs[idx] = e / s;
}

// ---- 2) fused edge scatter: deg, sum_p, agg. One wave per edge, ------------
//          2 edges per wave with prefetch of the 2nd edge's gather rows.
__device__ __forceinline__ void edge_body(int s, int d, int lane,
                                          const float* __restrict__ probs,
                                          const float4* __restrict__ h4,
                                          float* __restrict__ sum_p,
                                          float* __restrict__ deg,
                                          float* __restrict__ agg) {
    float p = probs[(size_t)s * 32 + lane];                  // coalesced 128B row
    atomicAdd(&sum_p[(size_t)d * 32 + lane], p);             // contiguous atomics
    if (lane == 0) atomicAdd(&deg[d], 1.0f);
    float4 hv = h4[(size_t)s * 32 + lane];                   // coalesced 512B row
    float* ad = agg + (size_t)d * 128 + lane * 4;
    atomicAdd(ad + 0, hv.x);
    atomicAdd(ad + 1, hv.y);
    atomicAdd(ad + 2, hv.z);
    atomicAdd(ad + 3, hv.w);
}

__global__ void k_edge(const int* __restrict__ src, const int* __restrict__ dst,
                       const float* __restrict__ probs,
                       const float4* __restrict__ h4,
                       float* __restrict__ sum_p, float* __restrict__ deg,
                       float* __restrict__ agg, int E) {
    int wid  = (blockIdx.x * blockDim.x + threadIdx.x) >> 5;
    int lane = threadIdx.x & 31;
    long e0 = (long)wid * 2;
    if (e0 >= E) return;
    int s0 = src[e0], d0 = dst[e0];
    bool has1 = (e0 + 1) < E;
    int s1 = 0, d1 = 0;
    if (has1) {
        s1 = src[e0 + 1];
        d1 = dst[e0 + 1];
        // gfx1250 global_prefetch_b8: pull next edge's gather rows toward WGP
        __builtin_prefetch(&probs[(size_t)s1 * 32 + lane], 0, 0);
        __builtin_prefetch(&h4[(size_t)s1 * 32 + lane], 0, 0);
    }
    edge_body(s0, d0, lane, probs, h4, sum_p, deg, agg);
    if (has1) edge_body(s1, d1, lane, probs, h4, sum_p, deg, agg);
}

// ---- 3) per-node KL + entropy, plus global sum / sum-of-squares (f64) -------
__global__ void k_f(const float* __restrict__ probs,
                    const float* __restrict__ sum_p,
                    const float* __restrict__ deg,
                    float* __restrict__ f1, float* __restrict__ f2,
                    double* __restrict__ acc, int N) {
    __shared__ float sh1[8], sh2[8];
    int wib  = threadIdx.x >> 5;
    int wid  = (blockIdx.x * blockDim.x + threadIdx.x) >> 5;
    int lane = threadIdx.x & 31;
    float f1v = 0.f, f2v = 0.f;
    if (wid < N) {
        size_t idx = (size_t)wid * 32 + lane;
        float dg   = deg[wid];
        float p    = probs[idx];
        float mean = sum_p[idx] / fmaxf(dg, 1.0f);
        float lp   = __logf(p + LOG_EPS);
        float lm   = __logf(mean + LOG_EPS);
        float t1   = wave_sum32(p * (lp - lm));     // KL(self || mean-neighbor)
        float t2   = wave_sum32(-mean * lm);        // entropy of consensus
        bool has   = dg > 0.0f;
        f1v = has ? t1 : 0.0f;
        f2v = has ? t2 : 0.0f;
        if (lane == 0) { f1[wid] = f1v; f2[wid] = f2v; }
    }
    if (lane == 0) { sh1[wib] = f1v; sh2[wib] = f2v; }
    __syncthreads();
    if (threadIdx.x == 0) {
        double a = 0, aq = 0, b = 0, bq = 0;
        #pragma unroll
        for (int i = 0; i < 8; ++i) {
            double x = (double)sh1[i], y = (double)sh2[i];
            a += x; aq += x * x; b += y; bq += y * y;
        }
        atomicAdd(&acc[0], a);   // global_atomic_add_f64
        atomicAdd(&acc[1], aq);
        atomicAdd(&acc[2], b);
        atomicAdd(&acc[3], bq);
    }
}

// ---- 4) finalize LayerNorm stats (mu, rsigma) + fold in tau scalars ---------
__global__ void k_stats(const double* __restrict__ acc,
                        const float* __restrict__ tau1,
                        const float* __restrict__ tau2,
                        float* __restrict__ stats, int N) {
    if (threadIdx.x == 0 && blockIdx.x == 0) {
        double inv = 1.0 / (double)N;
        double mu1 = acc[0] * inv, mu2 = acc[2] * inv;
        double v1  = acc[1] * inv - mu1 * mu1;
        double v2  = acc[3] * inv - mu2 * mu2;
        stats[0] = (float)mu1;
        stats[1] = (float)(1.0 / sqrt(v1 + (double)LN_EPS));
        stats[2] = (float)mu2;
        stats[3] = (float)(1.0 / sqrt(v2 + (double)LN_EPS));
        stats[4] = tau1[0];
        stats[5] = tau2[0];
    }
}

// ---- 5) fused gating + epilogue: one wave per node ---------------------------
//   Each lane handles one float4 of the 128-dim feature row (32 lanes * 4).
//   The scalar gating math (2x v_exp_f32) is recomputed per lane -- free next
//   to the 1KB/node of feature traffic -- which kills the gate[] round-trip.
__global__ void k_gate_final(const float4* __restrict__ h4,
                             const float4* __restrict__ agg4,
                             const float* __restrict__ f1,
                             const float* __restrict__ f2,
                             const float* __restrict__ old_z,
                             const float* __restrict__ norm,
                             const float* __restrict__ stats,
                             float4* __restrict__ out4,
                             float* __restrict__ out_z, int N) {
    int wid  = (blockIdx.x * blockDim.x + threadIdx.x) >> 5;
    int lane = threadIdx.x & 31;
    if (wid >= N) return;
    // wave-uniform per-node scalars
    float a = (f1[wid] - stats[0]) * stats[1] - stats[4];
    float b = (f2[wid] - stats[2]) * stats[3] - stats[5];
    // sigmoid(-(x - tau)) = 1 / (1 + e^(x - tau))
    float z = (1.0f / (1.0f + __expf(a))) * (1.0f / (1.0f + __expf(b)));
    float g = fminf(old_z[wid], z);
    float nm = norm[wid];
    if (lane == 0) out_z[wid] = z;

    size_t i = (size_t)wid * 32 + lane;      // float4 index into [N, 128]
    float4 av = agg4[i], hv = h4[i], r;
    r.x = hv.x + g * fmaxf(av.x * nm, 0.0f);
    r.y = hv.y + g * fmaxf(av.y * nm, 0.0f);
    r.z = hv.z + g * fmaxf(av.z * nm, 0.0f);
    r.w = hv.w + g * fmaxf(av.w * nm, 0.0f);
    out4[i] = r;
}

extern "C" void kernel_launch(void* const* d_in, const int* in_sizes, int n_in,
                              void* d_out, int out_size, void* d_ws, size_t ws_size,
                              hipStream_t stream) {
    const float* h      = (const float*)d_in[0];   // [N,128]
    const float* logits = (const float*)d_in[1];   // [N,32]
    const float* old_z  = (const float*)d_in[2];   // [N]
    const float* norm   = (const float*)d_in[3];   // [N,1]
    const float* tau_1  = (const float*)d_in[4];   // [1]
    const float* tau_2  = (const float*)d_in[5];   // [1]
    const int*   src    = (const int*)d_in[6];     // [E]
    const int*   dst    = (const int*)d_in[7];     // [E]

    const int N = in_sizes[2];
    const int E = in_sizes[6];
    const int C = 32;    // == wave32: one lane per class
    const int D = 128;   // 32 float4 per node

    // Workspace layout (accumulation region first so one memset covers it):
    //   [0,32)   : double acc[4]   (sum f1, sum f1^2, sum f2, sum f2^2)
    //   [32,64)  : float  stats[8] (mu1, rs1, mu2, rs2, tau1, tau2)
    //   [64, ..) : sum_p [N*C] | deg [N] | agg [N*D]   <- zeroed each call
    //   then     : probs [N*C] | f1 [N] | f2 [N]       (fully overwritten)
    char* base    = (char*)d_ws;
    double* acc   = (double*)base;
    float* stats  = (float*)(base + 32);
    float* sum_p  = (float*)(base + 64);
    float* deg    = sum_p + (size_t)N * C;
    float* agg    = deg + N;
    float* probs  = agg + (size_t)N * D;
    float* f1     = probs + (size_t)N * C;
    float* f2     = f1 + N;

    float* out_newh = (float*)d_out;                   // [N*128]
    float* out_z    = out_newh + (size_t)N * D;        // [N]

    const int BLK = 256;

    // 0) zero acc + stats + sum_p + deg + agg
    {
        size_t zero_bytes = 64 + ((size_t)N * C + N + (size_t)N * D) * sizeof(float);
        int n4 = (int)(zero_bytes / 16);
        k_zero<<<(n4 + BLK - 1) / BLK, BLK, 0, stream>>>((float4*)d_ws, n4);
    }
    // 1) softmax: N waves
    {
        long threads = (long)N * 32;
        k_softmax<<<(threads + BLK - 1) / BLK, BLK, 0, stream>>>(logits, probs, N);
    }
    // 2) edge scatter: ceil(E/2) waves, 2 edges per wave
    {
        long waves   = ((long)E + 1) / 2;
        long threads = waves * 32;
        k_edge<<<(threads + BLK - 1) / BLK, BLK, 0, stream>>>(
            src, dst, probs, (const float4*)h, sum_p, deg, agg, E);
    }
    // 3) per-node f1/f2 + global f64 reduction: N waves
    {
        long threads = (long)N * 32;
        k_f<<<(threads + BLK - 1) / BLK, BLK, 0, stream>>>(
            probs, sum_p, deg, f1, f2, acc, N);
    }
    // 4) stats finalize
    k_stats<<<1, 32, 0, stream>>>(acc, tau_1, tau_2, stats, N);
    // 5) fused gating + epilogue: N waves
    {
        long threads = (long)N * 32;
        k_gate_final<<<(threads + BLK - 1) / BLK, BLK, 0, stream>>>(
            (const float4*)h, (const float4*)agg, f1, f2, old_z, norm, stats,
            (float4*)out_newh, out_z, N);
    }
}